// EncoderRoGCN_12644383719569
// MI455X (gfx1250) — compile-verified
//
#include <hip/hip_runtime.h>

typedef __attribute__((ext_vector_type(2))) float v2f;
typedef __attribute__((ext_vector_type(8))) float v8f;

#define NDIM 128

// ---------------- zero fill (graph-capture safe, no hipMemsetAsync) --------
__global__ void rg_zero_kernel(float4* __restrict__ p, int n4) {
    int i = blockIdx.x * blockDim.x + threadIdx.x;
    if (i < n4) p[i] = make_float4(0.f, 0.f, 0.f, 0.f);
}

// ---------------- degree accumulation --------------------------------------
__global__ void rg_deg_kernel(const int* __restrict__ src, const int* __restrict__ dst,
                              float* __restrict__ odeg, float* __restrict__ ideg, int E) {
    int e = blockIdx.x * blockDim.x + threadIdx.x;
    if (e >= E) return;
    atomicAdd(&odeg[src[e]], 1.0f);
    atomicAdd(&ideg[dst[e]], 1.0f);
}

// ---------------- per-node normalization factors ---------------------------
// onorm = clip(out_deg,1)^-0.5 ; s1 = clip(in_deg,1)^-0.5 ; s2 = clip(in_deg,1)^-1.5
__global__ void rg_norm_kernel(const float* __restrict__ odeg, const float* __restrict__ ideg,
                               float* __restrict__ onorm, float* __restrict__ s1,
                               float* __restrict__ s2, int N) {
    int i = blockIdx.x * blockDim.x + threadIdx.x;
    if (i >= N) return;
    float od = fmaxf(odeg[i], 1.0f);
    float id = fmaxf(ideg[i], 1.0f);
    onorm[i] = rsqrtf(od);
    float r = rsqrtf(id);
    s1[i] = r;
    s2[i] = r / id;
}

// ---------------- edge scatter: mix[dst] += h[src]*onorm[src]*weight[etype] -
// One wave per edge; each lane owns 4 contiguous floats (float4 coalesced).
__global__ __launch_bounds__(256) void
rg_edge_kernel(const float* __restrict__ h, const float* __restrict__ onorm,
               const float* __restrict__ weight, const int* __restrict__ src,
               const int* __restrict__ dst, const int* __restrict__ etype,
               float* __restrict__ mix, int E) {
    int e = blockIdx.x * 8 + (threadIdx.x >> 5);
    if (e >= E) return;
    int lane = threadIdx.x & 31;
    int s = src[e];
    int d = dst[e];
    int t = etype[e];
    float sc = onorm[s];
    float4 hv = ((const float4*)(h + (size_t)s * NDIM))[lane];
    float4 wv = ((const float4*)(weight + (size_t)t * NDIM))[lane];
    float* mp = mix + (size_t)d * NDIM + lane * 4;
    atomicAdd(mp + 0, hv.x * wv.x * sc);
    atomicAdd(mp + 1, hv.y * wv.y * sc);
    atomicAdd(mp + 2, hv.z * wv.z * sc);
    atomicAdd(mp + 3, hv.w * wv.w * sc);
}

// ---------------- fused GEMM + epilogue ------------------------------------
// out[m,n] = (sum_k H[m,k]*W[k,n]) * s1[m] + mix[m,n] * s2[m]
// Block: 256 threads = 8 waves. Block handles 16 rows (one M tile), wave w
// handles columns [16w, 16w+16). K=128 swept in steps of 4 with
// v_wmma_f32_16x16x4_f32. H tile staged in LDS (stride 132 to avoid bank
// conflicts); W (64 KB) stays L2/L0-hot across all 3125 blocks.
#define LDS_STRIDE 132
__global__ __launch_bounds__(256) void
rg_gemm_kernel(const float* __restrict__ H, const float* __restrict__ W,
               const float* __restrict__ mix, const float* __restrict__ s1,
               const float* __restrict__ s2, float* __restrict__ out) {
    __shared__ float ldsH[16 * LDS_STRIDE];

    const int m0 = blockIdx.x * 16;
    const int tid = threadIdx.x;

    // cooperative load of 16x128 H tile (512 float4 / 256 threads)
    for (int i = tid; i < 16 * 32; i += 256) {
        int r = i >> 5;
        int c4 = i & 31;
        float4 v = ((const float4*)(H + (size_t)(m0 + r) * NDIM))[c4];
        float* dl = ldsH + r * LDS_STRIDE + c4 * 4;
        dl[0] = v.x; dl[1] = v.y; dl[2] = v.z; dl[3] = v.w;
    }
    __syncthreads();

    const int lane = tid & 31;
    const int hi = lane >> 4;      // half-wave: 0 or 1
    const int lo = lane & 15;
    const int n0 = (tid >> 5) * 16;  // wave's column tile

    v8f c = {};
    #pragma unroll
    for (int k = 0; k < NDIM; k += 4) {
        int kk = k + 2 * hi;
        // A 16x4 frag: row = lo, K = kk, kk+1
        v2f a;
        a.x = ldsH[lo * LDS_STRIDE + kk];
        a.y = ldsH[lo * LDS_STRIDE + kk + 1];
        // B 4x16 frag: col = n0+lo, K = kk, kk+1
        v2f b;
        b.x = W[(size_t)kk * NDIM + n0 + lo];
        b.y = W[(size_t)(kk + 1) * NDIM + n0 + lo];
        c = __builtin_amdgcn_wmma_f32_16x16x4_f32(
                /*neg_a=*/false, a, /*neg_b=*/false, b,
                /*c_mod=*/(short)0, c, /*reuse_a=*/false, /*reuse_b=*/false);
    }

    // C/D layout: VGPR g, lanes 0-15 -> row m0+g, lanes 16-31 -> row m0+8+g
    #pragma unroll
    for (int g = 0; g < 8; ++g) {
        int row = m0 + g + 8 * hi;
        float a1 = s1[row];
        float a2 = s2[row];
        size_t off = (size_t)row * NDIM + n0 + lo;
        out[off] = c[g] * a1 + mix[off] * a2;
    }
}

extern "C" void kernel_launch(void* const* d_in, const int* in_sizes, int n_in,
                              void* d_out, int out_size, void* d_ws, size_t ws_size,
                              hipStream_t stream) {
    const float* x      = (const float*)d_in[0];
    const float* weight = (const float*)d_in[1];
    const float* lw0    = (const float*)d_in[2];
    const float* lw1    = (const float*)d_in[3];
    const int*   src    = (const int*)d_in[4];
    const int*   dst    = (const int*)d_in[5];
    const int*   etype  = (const int*)d_in[6];
    float* out = (float*)d_out;

    const int N = in_sizes[0] / NDIM;   // 50000
    const int E = in_sizes[4];          // 600000
    const size_t ND = (size_t)N * NDIM;

    float* ws    = (float*)d_ws;
    float* h1    = ws;            // N*D
    float* mix   = h1 + ND;       // N*D
    float* odeg  = mix + ND;      // N
    float* ideg  = odeg + N;      // N
    float* onorm = ideg + N;      // N
    float* s1    = onorm + N;     // N
    float* s2    = s1 + N;        // N

    const int ZT = 256;
    // zero degree buffers (2*N floats, contiguous)
    {
        int n4 = (2 * N + 3) / 4;
        rg_zero_kernel<<<(n4 + ZT - 1) / ZT, ZT, 0, stream>>>((float4*)odeg, n4);
    }
    rg_deg_kernel<<<(E + ZT - 1) / ZT, ZT, 0, stream>>>(src, dst, odeg, ideg, E);
    rg_norm_kernel<<<(N + ZT - 1) / ZT, ZT, 0, stream>>>(odeg, ideg, onorm, s1, s2, N);

    const int edgeBlocks = (E + 7) / 8;
    const int mix4 = (int)(ND / 4);
    const int gemmBlocks = N / 16;   // 50000/16 = 3125 exact

    // ---- layer 1: h = x, loop_w = lw0 -> h1
    rg_zero_kernel<<<(mix4 + ZT - 1) / ZT, ZT, 0, stream>>>((float4*)mix, mix4);
    rg_edge_kernel<<<edgeBlocks, 256, 0, stream>>>(x, onorm, weight, src, dst, etype, mix, E);
    rg_gemm_kernel<<<gemmBlocks, 256, 0, stream>>>(x, lw0, mix, s1, s2, h1);

    // ---- layer 2: h = h1, loop_w = lw1 -> out
    rg_zero_kernel<<<(mix4 + ZT - 1) / ZT, ZT, 0, stream>>>((float4*)mix, mix4);
    rg_edge_kernel<<<edgeBlocks, 256, 0, stream>>>(h1, onorm, weight, src, dst, etype, mix, E);
    rg_gemm_kernel<<<gemmBlocks, 256, 0, stream>>>(h1, lw1, mix, s1, s2, out);
}